// GAT_67534065762746
// MI455X (gfx1250) — compile-verified
//
#include <hip/hip_runtime.h>
#include <cstdint>

#define NN 10000
#define EE 320000
#define ET (EE + NN)          // edges + self loops
#define DIN 256
#define DH 512                // 8 heads * 64
#define DOUT 32
#define HEADS 8
#define SLOPE 0.2f
#define WPB 8                 // waves per block
#define TPB 256

typedef __attribute__((ext_vector_type(2))) float v2f;
typedef __attribute__((ext_vector_type(8))) float v8f;

// order-preserving float<->uint encoding for atomicMax (handles negatives, -inf)
__device__ __forceinline__ unsigned fenc(float f) {
  unsigned u = __float_as_uint(f);
  return (u & 0x80000000u) ? ~u : (u | 0x80000000u);
}
__device__ __forceinline__ float fdec(unsigned u) {
  return __uint_as_float((u & 0x80000000u) ? (u & 0x7fffffffu) : ~u);
}

__global__ void k_init_u32(unsigned* __restrict__ p, int n, unsigned v) {
  int i = blockIdx.x * blockDim.x + threadIdx.x;
  if (i < n) p[i] = v;
}

__device__ __forceinline__ v8f wmma4(v2f a, v2f b, v8f c) {
  return __builtin_amdgcn_wmma_f32_16x16x4_f32(false, a, false, b, (short)0, c,
                                               false, false);
}

// ---------------- Fused dual WMMA f32 GEMM ----------------
// C0 = A @ B0, C1 = A @ B1 (row-major, MxK @ KxN). One wave computes a 16x32
// tile of BOTH outputs: the A fragment is loaded once and feeds 4 WMMAs per
// 4-wide k-step. N, K compile-time -> immediate-offset B addressing, no tail.
template <int N, int K>
__global__ void k_gemm_dual(const float* __restrict__ A,
                            const float* __restrict__ B0, const float* __restrict__ B1,
                            float* __restrict__ C0, float* __restrict__ C1, int M) {
  const int lane = threadIdx.x & 31;
  constexpr int TN = N >> 5;                 // 32-wide n-tiles
  const int tile = blockIdx.x * WPB + (threadIdx.x >> 5);
  if (tile >= (M >> 4) * TN) return;
  const int m0 = (tile / TN) << 4;
  const int n0 = (tile % TN) << 5;
  const int half = lane >> 4;                // 0: K pair {0,1}; 1: K pair {2,3}
  const int l15 = lane & 15;

  const float* arow = A + (size_t)(m0 + l15) * K + 2 * half;
  const float* b0 = B0 + (size_t)(2 * half) * N + (n0 + l15);
  const float* b1 = B1 + (size_t)(2 * half) * N + (n0 + l15);

  v8f acc00 = {}, acc01 = {}, acc10 = {}, acc11 = {};
#pragma unroll 4
  for (int k = 0; k < K; k += 4) {
    v2f a = *(const v2f*)arow;               // 8B aligned (K mult of 4, +2*half even)
    v2f u0, u1, w0, w1;
    u0.x = b0[0];      u0.y = b0[N];
    u1.x = b0[16];     u1.y = b0[N + 16];
    w0.x = b1[0];      w0.y = b1[N];
    w1.x = b1[16];     w1.y = b1[N + 16];
    acc00 = wmma4(a, u0, acc00);
    acc01 = wmma4(a, u1, acc01);
    acc10 = wmma4(a, w0, acc10);
    acc11 = wmma4(a, w1, acc11);
    arow += 4;
    b0 += (size_t)4 * N;
    b1 += (size_t)4 * N;
  }
  float* c0 = C0 + (size_t)(m0 + 8 * half) * N + (n0 + l15);
  float* c1 = C1 + (size_t)(m0 + 8 * half) * N + (n0 + l15);
#pragma unroll
  for (int r = 0; r < 8; ++r) {
    c0[(size_t)r * N] = acc00[r];
    c0[(size_t)r * N + 16] = acc01[r];
    c1[(size_t)r * N] = acc10[r];
    c1[(size_t)r * N + 16] = acc11[r];
  }
}

// ---------------- Layer 1 edge kernels (8 heads x 64 ch) ----------------
__device__ __forceinline__ void edge_sd(const long long* ei, int e, long long& s, long long& d) {
  if (e < EE) { s = ei[e]; d = ei[EE + e]; }
  else { s = e - EE; d = s; }
}

// pass A: alpha[e,h] = dot(leakyrelu(xl[s]+xr[d]), att[h]); atomic segment-max
__global__ void k_attn1(const float* __restrict__ xl, const float* __restrict__ xr,
                        const long long* __restrict__ ei, const float* __restrict__ att,
                        float* __restrict__ alpha, unsigned* __restrict__ amax) {
  const int lane = threadIdx.x & 31;
  const int e = blockIdx.x * WPB + (threadIdx.x >> 5);
  if (e >= ET) return;
  long long s, d; edge_sd(ei, e, s, d);
  const float* pl = xl + (size_t)s * DH + lane * 16;   // lane owns 16 contiguous ch
  const float* pr = xr + (size_t)d * DH + lane * 16;   // head = lane>>2
  const float* pa = att + lane * 16;
  float sum = 0.f;
#pragma unroll
  for (int i = 0; i < 16; ++i) {
    float v = pl[i] + pr[i];
    v = v > 0.f ? v : SLOPE * v;
    sum += v * pa[i];
  }
  sum += __shfl_xor(sum, 1, 32);
  sum += __shfl_xor(sum, 2, 32);
  if ((lane & 3) == 0) {
    int h = lane >> 2;
    alpha[(size_t)e * HEADS + h] = sum;
    atomicMax(&amax[(size_t)d * HEADS + h], fenc(sum));
  }
}

// pass B: ex = exp(alpha - max[dst]); denom[dst] += ex
__global__ void k_exp1(float* __restrict__ alpha, const unsigned* __restrict__ amax,
                       float* __restrict__ den, const long long* __restrict__ ei) {
  int idx = blockIdx.x * blockDim.x + threadIdx.x;
  if (idx >= ET * HEADS) return;
  int e = idx >> 3, h = idx & 7;
  long long d = (e < EE) ? ei[EE + e] : (long long)(e - EE);
  float ex = expf(alpha[idx] - fdec(amax[(size_t)d * HEADS + h]));
  alpha[idx] = ex;
  unsafeAtomicAdd(&den[(size_t)d * HEADS + h], ex);
}

// pass C: agg[dst] += xl[src] * (ex / denom[dst])
__global__ void k_agg1(const float* __restrict__ xl, const float* __restrict__ alpha,
                       const float* __restrict__ den, const long long* __restrict__ ei,
                       float* __restrict__ agg) {
  const int lane = threadIdx.x & 31;
  const int e = blockIdx.x * WPB + (threadIdx.x >> 5);
  if (e >= ET) return;
  long long s, d; edge_sd(ei, e, s, d);
  const int h = lane >> 2;
  const float coeff = alpha[(size_t)e * HEADS + h] / (den[(size_t)d * HEADS + h] + 1e-16f);
  const float* pl = xl + (size_t)s * DH + lane * 16;
  float* pd = agg + (size_t)d * DH + lane * 16;
#pragma unroll
  for (int i = 0; i < 16; ++i) unsafeAtomicAdd(&pd[i], pl[i] * coeff);
}

// h = elu(agg + b1), in place
__global__ void k_elu_bias(float* __restrict__ a, const float* __restrict__ b, int total) {
  int i = blockIdx.x * blockDim.x + threadIdx.x;
  if (i >= total) return;
  float v = a[i] + b[i & (DH - 1)];
  a[i] = v > 0.f ? v : expf(v) - 1.f;
}

// ---------------- Layer 2 edge kernels (1 head x 32 ch) ----------------
__global__ void k_attn2(const float* __restrict__ hl, const float* __restrict__ hr,
                        const long long* __restrict__ ei, const float* __restrict__ att,
                        float* __restrict__ alpha, unsigned* __restrict__ amax) {
  const int lane = threadIdx.x & 31;
  const int e = blockIdx.x * WPB + (threadIdx.x >> 5);
  if (e >= ET) return;
  long long s, d; edge_sd(ei, e, s, d);
  float v = hl[(size_t)s * DOUT + lane] + hr[(size_t)d * DOUT + lane];
  v = v > 0.f ? v : SLOPE * v;
  float sum = v * att[lane];
#pragma unroll
  for (int o = 16; o; o >>= 1) sum += __shfl_xor(sum, o, 32);
  if (lane == 0) {
    alpha[e] = sum;
    atomicMax(&amax[d], fenc(sum));
  }
}

__global__ void k_exp2(float* __restrict__ alpha, const unsigned* __restrict__ amax,
                       float* __restrict__ den, const long long* __restrict__ ei) {
  int e = blockIdx.x * blockDim.x + threadIdx.x;
  if (e >= ET) return;
  long long d = (e < EE) ? ei[EE + e] : (long long)(e - EE);
  float ex = expf(alpha[e] - fdec(amax[d]));
  alpha[e] = ex;
  unsafeAtomicAdd(&den[d], ex);
}

__global__ void k_agg2(const float* __restrict__ hl, const float* __restrict__ alpha,
                       const float* __restrict__ den, const long long* __restrict__ ei,
                       float* __restrict__ agg) {
  const int lane = threadIdx.x & 31;
  const int e = blockIdx.x * WPB + (threadIdx.x >> 5);
  if (e >= ET) return;
  long long s, d; edge_sd(ei, e, s, d);
  float coeff = alpha[e] / (den[d] + 1e-16f);
  unsafeAtomicAdd(&agg[(size_t)d * DOUT + lane], hl[(size_t)s * DOUT + lane] * coeff);
}

// out = log_softmax(agg + b2) — one wave per row (32 lanes == 32 cols)
__global__ void k_out(const float* __restrict__ agg, const float* __restrict__ b,
                      float* __restrict__ out) {
  const int lane = threadIdx.x & 31;
  const int n = blockIdx.x * WPB + (threadIdx.x >> 5);
  if (n >= NN) return;
  float v = agg[(size_t)n * DOUT + lane] + b[lane];
  float m = v;
#pragma unroll
  for (int o = 16; o; o >>= 1) m = fmaxf(m, __shfl_xor(m, o, 32));
  float ex = expf(v - m), sum = ex;
#pragma unroll
  for (int o = 16; o; o >>= 1) sum += __shfl_xor(sum, o, 32);
  out[(size_t)n * DOUT + lane] = v - m - logf(sum);
}

// ---------------------------------------------------------------------------------
extern "C" void kernel_launch(void* const* d_in, const int* in_sizes, int n_in,
                              void* d_out, int out_size, void* d_ws, size_t ws_size,
                              hipStream_t stream) {
  (void)in_sizes; (void)n_in; (void)out_size; (void)ws_size;
  const float*     x    = (const float*)d_in[0];
  const long long* ei   = (const long long*)d_in[1];
  const float*     W1l  = (const float*)d_in[2];
  const float*     W1r  = (const float*)d_in[3];
  const float*     att1 = (const float*)d_in[4];
  const float*     b1   = (const float*)d_in[5];
  const float*     W2l  = (const float*)d_in[6];
  const float*     W2r  = (const float*)d_in[7];
  const float*     att2 = (const float*)d_in[8];
  const float*     b2   = (const float*)d_in[9];
  float* out = (float*)d_out;

  char* ws = (char*)d_ws;
  size_t off = 0;
  auto take = [&](size_t bytes) {
    void* p = ws + off;
    off += (bytes + 255) & ~(size_t)255;
    return p;
  };
  float*    XL1   = (float*)take((size_t)NN * DH * 4);      // 20.5 MB
  float*    XR1   = (float*)take((size_t)NN * DH * 4);
  float*    AGG1  = (float*)take((size_t)NN * DH * 4);      // reused as H after ELU
  float*    ALPH1 = (float*)take((size_t)ET * HEADS * 4);
  unsigned* AMAX1 = (unsigned*)take((size_t)NN * HEADS * 4);
  float*    DEN1  = (float*)take((size_t)NN * HEADS * 4);
  float*    HL2   = (float*)take((size_t)NN * DOUT * 4);
  float*    HR2   = (float*)take((size_t)NN * DOUT * 4);
  float*    AGG2  = (float*)take((size_t)NN * DOUT * 4);
  float*    ALPH2 = (float*)take((size_t)ET * 4);
  unsigned* AMAX2 = (unsigned*)take((size_t)NN * 4);
  float*    DEN2  = (float*)take((size_t)NN * 4);           // total ~78 MB

  // zero accumulators each call (harness poisons ws once, never restores)
  hipMemsetAsync(AGG1, 0, (size_t)NN * DH * 4, stream);
  hipMemsetAsync(DEN1, 0, (size_t)NN * HEADS * 4, stream);
  hipMemsetAsync(AGG2, 0, (size_t)NN * DOUT * 4, stream);
  hipMemsetAsync(DEN2, 0, (size_t)NN * 4, stream);
  const unsigned NEG_INF_ENC = 0x007FFFFFu;                 // fenc(-inf)
  k_init_u32<<<(NN * HEADS + 255) / 256, 256, 0, stream>>>(AMAX1, NN * HEADS, NEG_INF_ENC);
  k_init_u32<<<(NN + 255) / 256, 256, 0, stream>>>(AMAX2, NN, NEG_INF_ENC);

  // layer 1 fused dual GEMM: XL1/XR1 = x @ {W1l, W1r}  (10000x256 @ 256x512)
  {
    int tiles = (NN / 16) * (DH / 32);
    k_gemm_dual<DH, DIN><<<(tiles + WPB - 1) / WPB, TPB, 0, stream>>>(
        x, W1l, W1r, XL1, XR1, NN);
  }
  int egrid = (ET + WPB - 1) / WPB;
  k_attn1<<<egrid, TPB, 0, stream>>>(XL1, XR1, ei, att1, ALPH1, AMAX1);
  k_exp1<<<(ET * HEADS + TPB - 1) / TPB, TPB, 0, stream>>>(ALPH1, AMAX1, DEN1, ei);
  k_agg1<<<egrid, TPB, 0, stream>>>(XL1, ALPH1, DEN1, ei, AGG1);
  k_elu_bias<<<(NN * DH + TPB - 1) / TPB, TPB, 0, stream>>>(AGG1, b1, NN * DH);

  // layer 2 fused dual GEMM: HL2/HR2 = h @ {W2l, W2r}  (10000x512 @ 512x32)
  {
    int tiles = (NN / 16) * (DOUT / 32);
    k_gemm_dual<DOUT, DH><<<(tiles + WPB - 1) / WPB, TPB, 0, stream>>>(
        AGG1, W2l, W2r, HL2, HR2, NN);
  }
  k_attn2<<<egrid, TPB, 0, stream>>>(HL2, HR2, ei, att2, ALPH2, AMAX2);
  k_exp2<<<(ET + TPB - 1) / TPB, TPB, 0, stream>>>(ALPH2, AMAX2, DEN2, ei);
  k_agg2<<<egrid, TPB, 0, stream>>>(HL2, ALPH2, DEN2, ei, AGG2);
  k_out<<<(NN + WPB - 1) / WPB, TPB, 0, stream>>>(AGG2, b2, out);
}